// ModelNew_50379966382552
// MI455X (gfx1250) — compile-verified
//
#include <hip/hip_runtime.h>

typedef __attribute__((ext_vector_type(2))) float v2f;
typedef __attribute__((ext_vector_type(8))) float v8f;

#define FEAT 64

// One wave (32 lanes) handles a 16-destination-row block x 32 features,
// accumulating 4 edges per step with V_WMMA_F32_16X16X4_F32:
//   D(16x16) += A(16x4 edge-weights) * B(4x16 gathered features)
// Two waves per 16-row block (feature halves). CSR rows are contiguous, so
// the block's edges are one contiguous range [row_ptr[m0], row_ptr[m0+16]).
__global__ __launch_bounds__(256)
void gcn_agg_wmma_kernel(const int* __restrict__ row_ptr,
                         const int* __restrict__ col_idx,
                         const float* __restrict__ feat,
                         const float* __restrict__ degrees,
                         float* __restrict__ out,
                         int n_nodes, int n_edges)
{
    const int lane        = threadIdx.x & 31;
    const int waveInBlock = threadIdx.x >> 5;            // 0..7
    const int gwave       = blockIdx.x * 8 + waveInBlock;
    const int nBlocks     = (n_nodes + 15) >> 4;
    const int blk         = gwave >> 1;                  // 16-row block id
    const int half        = gwave & 1;                   // which 32-feature half
    if (blk >= nBlocks) return;                          // wave-uniform exit

    const int m0  = blk << 4;
    const int m   = lane & 15;                           // A/C row index (both half-waves)
    const int row = m0 + m;

    // Per-lane CSR bounds for "does edge e belong to my row" test.
    int   rp0 = 0, rp1 = 0;
    float degd = 1.0f;
    if (row < n_nodes) {
        rp0  = row_ptr[row];
        rp1  = row_ptr[row + 1];
        degd = degrees[row];
    }

    // Wave-uniform edge range of the whole 16-row block. Force into SGPRs so
    // the edge loop is scalar (s_cbranch) -> EXEC stays all-1s at every WMMA.
    const int m_hi    = (m0 + 16 < n_nodes) ? (m0 + 16) : n_nodes;
    const int e_start = __builtin_amdgcn_readfirstlane(row_ptr[m0]);
    const int e_end   = __builtin_amdgcn_readfirstlane(row_ptr[m_hi]);

    // A-matrix lane layout (32-bit A 16x4): lanes 0-15 supply K=0,1 in v0,v1;
    // lanes 16-31 supply K=2,3. B (4x16) mirrors: lane n (&15) holds row K=kA
    // in v0 and K=kA+1 in v1.
    const int kA    = (lane < 16) ? 0 : 2;
    const int nIn   = lane & 15;                         // feature col within tile
    const int nBase = (half << 5) + nIn;                 // tile0 col; tile1 = +16
    const int eClamp = n_edges - 1;

    v8f acc0 = {};   // features nBase
    v8f acc1 = {};   // features nBase+16

#pragma unroll 2
    for (int e = e_start; e < e_end; e += 4) {
        const int e0 = e + kA;
        const int e1 = e0 + 1;
        // Clamp indices for safe loads; out-of-row contributions get weight 0.
        const int ec0 = (e0 < eClamp) ? e0 : eClamp;
        const int ec1 = (e1 < eClamp) ? e1 : eClamp;
        const int s0  = col_idx[ec0];
        const int s1  = col_idx[ec1];

        // Unconditional gather + rsqrt (always safe: s0/s1 are valid node ids),
        // then branch-free select -> v_cndmask, no exec-mask regions.
        const float rs0 = __frsqrt_rn(degd * degrees[s0]);
        const float rs1 = __frsqrt_rn(degd * degrees[s1]);
        const bool  in0 = (e0 >= rp0) & (e0 < rp1);
        const bool  in1 = (e1 >= rp0) & (e1 < rp1);
        v2f a;
        a.x = in0 ? rs0 : 0.0f;
        a.y = in1 ? rs1 : 0.0f;

        const float* r0 = feat + s0 * FEAT;
        const float* r1 = feat + s1 * FEAT;
        v2f b0; b0.x = r0[nBase];      b0.y = r1[nBase];
        v2f b1; b1.x = r0[nBase + 16]; b1.y = r1[nBase + 16];

        acc0 = __builtin_amdgcn_wmma_f32_16x16x4_f32(
                   false, a, false, b0, (short)0, acc0, false, false);
        acc1 = __builtin_amdgcn_wmma_f32_16x16x4_f32(
                   false, a, false, b1, (short)0, acc1, false, false);
    }

    // C/D layout: VGPR r -> M = r (lanes 0-15) or 8+r (lanes 16-31); N = lane&15.
    const int mBase = m0 + ((lane < 16) ? 0 : 8);
    const int cBase = (half << 5) + nIn;
    if (m0 + 16 <= n_nodes) {
        // Whole block in range: unconditional stores.
#pragma unroll
        for (int r = 0; r < 8; ++r) {
            const int orow = mBase + r;
            out[orow * FEAT + cBase]      = acc0[r];
            out[orow * FEAT + cBase + 16] = acc1[r];
        }
    } else {
#pragma unroll
        for (int r = 0; r < 8; ++r) {
            const int orow = mBase + r;
            if (orow < n_nodes) {
                out[orow * FEAT + cBase]      = acc0[r];
                out[orow * FEAT + cBase + 16] = acc1[r];
            }
        }
    }
}

extern "C" void kernel_launch(void* const* d_in, const int* in_sizes, int n_in,
                              void* d_out, int out_size, void* d_ws, size_t ws_size,
                              hipStream_t stream) {
    const int*   row_ptr = (const int*)d_in[0];
    const int*   col_idx = (const int*)d_in[1];
    const float* feat    = (const float*)d_in[2];
    const float* degs    = (const float*)d_in[3];
    float*       out     = (float*)d_out;

    const int n_nodes = in_sizes[0] - 1;   // row_ptr has N+1 entries
    const int n_edges = in_sizes[1];

    const int nBlocks = (n_nodes + 15) / 16;
    const int nWaves  = nBlocks * 2;                 // 2 waves (feature halves) per block
    const int grid    = (nWaves + 7) / 8;            // 8 waves per 256-thread workgroup

    gcn_agg_wmma_kernel<<<grid, 256, 0, stream>>>(
        row_ptr, col_idx, feat, degs, out, n_nodes, n_edges);
}